// NUFFT2D_Torch_80144089743688
// MI455X (gfx1250) — compile-verified
//
#include <hip/hip_runtime.h>
#include <hip/hip_bf16.h>
#include <math.h>

// ---------------------------------------------------------------------------
// NUFFT2D dir_op for MI455X (gfx1250):
//   prep:   apodize (÷scaling), centered zero-pad 512->1024, fold fftshift
//   fftA:   1024-pt row FFT per workgroup (radix 16*16*4; radix-16 stages as
//           fp32 WMMA GEMMs F16(16x16) x X), transposed store
//   fftB:   same kernel on transposed buffer == column FFT, transposed store
//   degrid: 300000 x 36-tap complex gather-MAC, output fftshift folded as
//           index XOR 0x80200, 1/K_NORM folded into epilogue;
//           idx/weight streams read as b128 vectors
// Workspace: 2 x 8 MB float2 grids in d_ws (needs >= 16 MB).
// ---------------------------------------------------------------------------

typedef float v2f __attribute__((ext_vector_type(2)));
typedef float v8f __attribute__((ext_vector_type(8)));

static __device__ __forceinline__ v8f wmma_f32_k4(v2f a, v2f b, v8f c) {
  // D = A(16x4,f32) * B(4x16,f32) + C(16x16,f32), exact fp32 FMA semantics.
  return __builtin_amdgcn_wmma_f32_16x16x4_f32(
      /*neg_a=*/false, a, /*neg_b=*/false, b,
      /*c_mod=*/(short)0, c, /*reuse_a=*/false, /*reuse_b=*/false);
}

#define KD   1024
#define ND   512
#define PADL 256
#define TWO_PI 6.28318530717958647692f

// ---------------- Stage 1: apodize + pad + input fftshift -------------------
__global__ void nufft_prep(const float* __restrict__ xr,
                           const float* __restrict__ xi,
                           const float* __restrict__ sc,
                           float2* __restrict__ g) {
  int idx = blockIdx.x * blockDim.x + threadIdx.x;
  if (idx >= KD * KD) return;
  int r = idx >> 10, c = idx & (KD - 1);
  int u = r ^ (KD / 2);            // fold fftshift of padded array (shift=512)
  int v = c ^ (KD / 2);
  float2 val; val.x = 0.f; val.y = 0.f;
  if (u >= PADL && u < PADL + ND && v >= PADL && v < PADL + ND) {
    int i = u - PADL, j = v - PADL;
    float inv = 1.0f / sc[i * ND + j];
    val.x = xr[i * ND + j] * inv;
    val.y = xi[i * ND + j] * inv;
  }
  g[idx] = val;
}

// ---------------- Stage 2/3: 1024-pt line FFT, WMMA radix-16 ----------------
// Four-step: n = n1*64 + n2 ; k = k2*16 + k1 ; k2 = j2*16 + j1 ; n2 = m1*4+m2
//   A: DFT16 over n1 (WMMA GEMM, 4 tiles of 16 n2-columns)
//   twiddle w1024^{k1*n2}
//   B: DFT16 over m1 (WMMA GEMM, 4 phases m2)
//   twiddle w64^{j1*m2}
//   C: DFT4 over m2 (VALU), transposed global store
__global__ void __launch_bounds__(256) nufft_fft_line(const float2* __restrict__ in,
                                                      float2* __restrict__ out) {
  __shared__ float Fr[256], Fi[256];            // DFT16 matrix e^{-2pi i mk/16}
  __shared__ float xre[1024], xim[1024];        // input, reused as z (stage B out)
  __shared__ float yre[1024], yim[1024];        // stage A out

  const int tid  = threadIdx.x;
  const int lane = tid & 31;
  const int wave = tid >> 5;
  const int row  = blockIdx.x;

  // DFT16 twiddle table
  {
    int m = tid >> 4, k = tid & 15;
    float ang = -(TWO_PI / 16.0f) * (float)(m * k);
    Fr[tid] = cosf(ang);
    Fi[tid] = sinf(ang);
  }
  // load line
#pragma unroll
  for (int j = 0; j < 4; ++j) {
    int p = tid + 256 * j;
    float2 v = in[row * KD + p];
    xre[p] = v.x; xim[p] = v.y;
  }
  __syncthreads();

  const int Mr  = lane & 15;                 // A-operand row / B-operand col
  const int hi2 = (lane >> 4) << 1;          // K offset for upper half-wave

  // ---- Stage A: Y[k1][n2] = sum_n1 F16[k1][n1] * X[n1*64 + n2] ----
  {
    const int tile = wave & 3;               // n2 block: n2 = 16*tile + Mr
    const int part = wave >> 2;              // 0 = real out, 1 = imag out
    const float* Bs1 = part ? xim : xre;     // paired with  Fr
    const float* Bs2 = part ? xre : xim;     // paired with +-Fi
    const float sgn  = part ? 1.0f : -1.0f;
    const int ncol = 16 * tile + Mr;
    v8f acc = {0.f, 0.f, 0.f, 0.f, 0.f, 0.f, 0.f, 0.f};
#pragma unroll
    for (int s = 0; s < 4; ++s) {
      int K = 4 * s + hi2;
      v2f a, b;
      a.x = Fr[Mr * 16 + K];      a.y = Fr[Mr * 16 + K + 1];
      b.x = Bs1[K * 64 + ncol];   b.y = Bs1[(K + 1) * 64 + ncol];
      acc = wmma_f32_k4(a, b, acc);
    }
#pragma unroll
    for (int s = 0; s < 4; ++s) {
      int K = 4 * s + hi2;
      v2f a, b;
      a.x = sgn * Fi[Mr * 16 + K];  a.y = sgn * Fi[Mr * 16 + K + 1];
      b.x = Bs2[K * 64 + ncol];     b.y = Bs2[(K + 1) * 64 + ncol];
      acc = wmma_f32_k4(a, b, acc);
    }
    float* dst = part ? yim : yre;
#pragma unroll
    for (int r = 0; r < 8; ++r)
      dst[(r + (hi2 << 2)) * 64 + ncol] = acc[r];   // k1 = r + 8*hi
  }
  __syncthreads();

  // ---- twiddle w1024^{k1*n2} ----
#pragma unroll
  for (int j = 0; j < 4; ++j) {
    int p = tid + 256 * j;
    int k1 = p >> 6, n2 = p & 63;
    float ang = -(TWO_PI / 1024.0f) * (float)(k1 * n2);
    float cw = cosf(ang), sw = sinf(ang);
    float tr = yre[p] * cw - yim[p] * sw;
    float ti = yre[p] * sw + yim[p] * cw;
    yre[p] = tr; yim[p] = ti;
  }
  __syncthreads();

  // ---- Stage B: Z[k1][j1][m2] = sum_m1 F16[j1][m1] * Y[k1][m1*4+m2] ----
  {
    const int m2   = wave & 3;
    const int part = wave >> 2;
    const float* Bs1 = part ? yim : yre;
    const float* Bs2 = part ? yre : yim;
    const float sgn  = part ? 1.0f : -1.0f;
    v8f acc = {0.f, 0.f, 0.f, 0.f, 0.f, 0.f, 0.f, 0.f};
#pragma unroll
    for (int s = 0; s < 4; ++s) {
      int K = 4 * s + hi2;                           // m1
      v2f a, b;
      a.x = Fr[Mr * 16 + K];             a.y = Fr[Mr * 16 + K + 1];
      b.x = Bs1[Mr * 64 + K * 4 + m2];   b.y = Bs1[Mr * 64 + (K + 1) * 4 + m2];
      acc = wmma_f32_k4(a, b, acc);
    }
#pragma unroll
    for (int s = 0; s < 4; ++s) {
      int K = 4 * s + hi2;
      v2f a, b;
      a.x = sgn * Fi[Mr * 16 + K];       a.y = sgn * Fi[Mr * 16 + K + 1];
      b.x = Bs2[Mr * 64 + K * 4 + m2];   b.y = Bs2[Mr * 64 + (K + 1) * 4 + m2];
      acc = wmma_f32_k4(a, b, acc);
    }
    float* dst = part ? xim : xre;                   // reuse x as z
#pragma unroll
    for (int r = 0; r < 8; ++r)
      dst[Mr * 64 + (r + (hi2 << 2)) * 4 + m2] = acc[r];  // j1 = r + 8*hi
  }
  __syncthreads();

  // ---- twiddle w64^{j1*m2} ----
#pragma unroll
  for (int j = 0; j < 4; ++j) {
    int p = tid + 256 * j;
    int rem = p & 63, j1 = rem >> 2, m2 = rem & 3;
    float ang = -(TWO_PI / 64.0f) * (float)(j1 * m2);
    float cw = cosf(ang), sw = sinf(ang);
    float tr = xre[p] * cw - xim[p] * sw;
    float ti = xre[p] * sw + xim[p] * cw;
    xre[p] = tr; xim[p] = ti;
  }
  __syncthreads();

  // ---- Stage C: radix-4 over m2, transposed global store ----
  {
    int k1 = tid >> 4, j1 = tid & 15;
    int base = k1 * 64 + j1 * 4;
    float z0r = xre[base + 0], z0i = xim[base + 0];
    float z1r = xre[base + 1], z1i = xim[base + 1];
    float z2r = xre[base + 2], z2i = xim[base + 2];
    float z3r = xre[base + 3], z3i = xim[base + 3];
    float e0r = z0r + z2r, e0i = z0i + z2i, d0r = z0r - z2r, d0i = z0i - z2i;
    float e1r = z1r + z3r, e1i = z1i + z3i, d1r = z1r - z3r, d1i = z1i - z3i;
    int kb = j1 * 16 + k1;
    float2 o;
    o.x = e0r + e1r; o.y = e0i + e1i; out[(kb        ) * KD + row] = o;  // j2=0
    o.x = d0r + d1i; o.y = d0i - d1r; out[(kb + 256  ) * KD + row] = o;  // j2=1 (-i)
    o.x = e0r - e1r; o.y = e0i - e1i; out[(kb + 512  ) * KD + row] = o;  // j2=2
    o.x = d0r - d1i; o.y = d0i + d1r; out[(kb + 768  ) * KD + row] = o;  // j2=3 (+i)
  }
}

// ---------------- Stage 4: de-gridding gather ------------------------------
// 36 taps = 9 x int4 index loads + 9 x float4 weight loads (rows are 144 B,
// so every row base is 16-byte aligned) + 36 random 8-B L2-resident gathers.
__global__ void nufft_degrid(const float2* __restrict__ kk,
                             const int* __restrict__ gidx,
                             const float* __restrict__ kbv,
                             float2* __restrict__ outp, int M) {
  int m = blockIdx.x * blockDim.x + threadIdx.x;
  if (m >= M) return;
  const int4*   gi4 = (const int4*)  (gidx) + (size_t)m * 9;
  const float4* kv4 = (const float4*)(kbv)  + (size_t)m * 9;
  float ar = 0.f, ai = 0.f;
#pragma unroll
  for (int t = 0; t < 9; ++t) {
    int4   g4 = gi4[t];
    float4 w4 = kv4[t];
    float2 v;
    v = kk[g4.x ^ 0x80200]; ar = fmaf(v.x, w4.x, ar); ai = fmaf(v.y, w4.x, ai);
    v = kk[g4.y ^ 0x80200]; ar = fmaf(v.x, w4.y, ar); ai = fmaf(v.y, w4.y, ai);
    v = kk[g4.z ^ 0x80200]; ar = fmaf(v.x, w4.z, ar); ai = fmaf(v.y, w4.z, ai);
    v = kk[g4.w ^ 0x80200]; ar = fmaf(v.x, w4.w, ar); ai = fmaf(v.y, w4.w, ai);
  }
  float2 o; o.x = ar * (1.0f / 1024.0f); o.y = ai * (1.0f / 1024.0f);
  outp[m] = o;
}

extern "C" void kernel_launch(void* const* d_in, const int* in_sizes, int n_in,
                              void* d_out, int out_size, void* d_ws, size_t ws_size,
                              hipStream_t stream) {
  const float* xx_real = (const float*)d_in[0];   // (1,512,512) f32
  const float* xx_imag = (const float*)d_in[1];   // (1,512,512) f32
  const float* scaling = (const float*)d_in[2];   // (512,512)   f32
  const float* kb_vals = (const float*)d_in[3];   // (M,36)      f32
  const int*   grididx = (const int*)  d_in[4];   // (M,36)      i32
  const int M = in_sizes[3] / 36;

  float2* buf0 = (float2*)d_ws;                   // 8 MB
  float2* buf1 = buf0 + (size_t)KD * KD;          // 8 MB

  nufft_prep<<<(KD * KD + 255) / 256, 256, 0, stream>>>(xx_real, xx_imag, scaling, buf0);
  nufft_fft_line<<<KD, 256, 0, stream>>>(buf0, buf1);  // rows,  transposed out
  nufft_fft_line<<<KD, 256, 0, stream>>>(buf1, buf0);  // cols,  natural order out
  nufft_degrid<<<(M + 255) / 256, 256, 0, stream>>>(buf0, grididx, kb_vals,
                                                    (float2*)d_out, M);
}